// GCN_48034914238866
// MI455X (gfx1250) — compile-verified
//
#include <hip/hip_runtime.h>

// ---------------------------------------------------------------------------
// GCN forward on MI455X (gfx1250), fp32 end-to-end.
//   per layer: H = X @ W  (V_WMMA_F32_16X16X4_F32, one wave per 16x16 tile)
//              A[dst] += (dinv[src]*dinv[dst]) * H[src]   (hw f32 atomics)
//              X' = relu(A + b)
//   classifier: out = X @ CW + CB  (40 cols, trivially small -> VALU kernel)
// ---------------------------------------------------------------------------

#define N_NODES 50000
#define N_EDGES 800000
#define F_IN    128
#define NCLS    40

typedef __attribute__((ext_vector_type(2))) float v2f;
typedef __attribute__((ext_vector_type(4))) float v4f;
typedef __attribute__((ext_vector_type(8))) float v8f;

// ---------------------------------------------------------------- utilities
__global__ void zero_f32(float* __restrict__ p, long long n) {
  long long i = (long long)blockIdx.x * blockDim.x + threadIdx.x;
  if (i < n) p[i] = 0.0f;
}

// deg[dst] += 1 for every edge (incl. self loops). deg buffer pre-zeroed.
__global__ void degree_kernel(const long long* __restrict__ ei,
                              float* __restrict__ deg) {
  int i = blockIdx.x * blockDim.x + threadIdx.x;
  if (i >= N_EDGES + N_NODES) return;
  int d = (i < N_EDGES) ? (int)ei[N_EDGES + i]   // edge_index[1][i]
                        : (i - N_EDGES);         // self loop
  unsafeAtomicAdd(&deg[d], 1.0f);
}

__global__ void rsqrt_kernel(float* __restrict__ deg) {
  int i = blockIdx.x * blockDim.x + threadIdx.x;
  if (i >= N_NODES) return;
  float d = deg[i];
  deg[i] = (d > 0.0f) ? rsqrtf(d) : 0.0f;
}

// ------------------------------------------------------------- fp32 WMMA GEMM
// H[N_NODES x DOUT] = X[N_NODES x DIN] * W[DIN x DOUT]
// One wave computes one 16x16 tile with V_WMMA_F32_16X16X4_F32 over K.
// N_NODES = 3125*16 exactly; DIN/DOUT multiples of 16 -> no ragged tiles.
// Excess tail waves are clamped onto the last tile (identical redundant work)
// so EXEC stays all-1s for every WMMA issue (ISA 7.12 requirement).
template <int DIN, int DOUT>
__global__ __launch_bounds__(256) void gemm_wmma_f32(
    const float* __restrict__ X, const float* __restrict__ W,
    float* __restrict__ H) {
  constexpr int MT    = N_NODES / 16;
  constexpr int NT    = DOUT / 16;
  constexpr int TOTAL = MT * NT;

  int wg = (int)((blockIdx.x * blockDim.x + threadIdx.x) >> 5);  // wave id
  if (wg >= TOTAL) wg = TOTAL - 1;  // uniform per wave; keeps EXEC = all 1s
  const int mt   = wg / NT;
  const int nt   = wg % NT;
  const int lane = threadIdx.x & 31;
  const int l16  = lane & 15;
  const int kh   = (lane >> 4) << 1;  // K sub-offset: 0 (lanes 0-15) / 2 (16-31)

  // A: lane holds X[mt*16 + l16][k + kh + {0,1}]  (contiguous -> 8B load)
  const float* __restrict__ arow = X + (long long)(mt * 16 + l16) * DIN + kh;
  // B: lane holds W[k + kh + {0,1}][nt*16 + l16]
  const float* __restrict__ bcol = W + (long long)kh * DOUT + (nt * 16 + l16);

  v8f acc = {};
#pragma unroll 8
  for (int k = 0; k < DIN; k += 4) {
    v2f a = *(const v2f*)(arow + k);
    v2f b;
    b.x = bcol[(long long)k * DOUT];
    b.y = bcol[(long long)(k + 1) * DOUT];
    acc = __builtin_amdgcn_wmma_f32_16x16x4_f32(
        /*neg_a=*/false, a, /*neg_b=*/false, b,
        /*c_mod=*/(short)0, acc, /*reuse_a=*/false, /*reuse_b=*/false);
  }

  // D layout: VGPR r -> row r (lanes 0-15) / row 8+r (lanes 16-31), col = l16
  float* __restrict__ out =
      H + (long long)(mt * 16 + (lane >> 4) * 8) * DOUT + nt * 16 + l16;
#pragma unroll
  for (int r = 0; r < 8; ++r) out[(long long)r * DOUT] = acc[r];
}

// ------------------------------------------------------- edge scatter-gather
// acc[dst] += dinv[src]*dinv[dst] * H[src], float4 per thread, f32 HW atomics.
template <int D>
__global__ void aggregate_kernel(const float* __restrict__ H,
                                 const long long* __restrict__ ei,
                                 const float* __restrict__ dinv,
                                 float* __restrict__ acc) {
  constexpr int V = D / 4;
  constexpr long long TOTAL = (long long)(N_EDGES + N_NODES) * V;
  long long idx = (long long)blockIdx.x * blockDim.x + threadIdx.x;
  if (idx >= TOTAL) return;
  int e = (int)(idx / V);
  int c = (int)(idx % V) * 4;
  int s, d;
  if (e < N_EDGES) {
    s = (int)ei[e];             // edge_index[0][e]
    d = (int)ei[N_EDGES + e];   // edge_index[1][e]
  } else {
    s = d = e - N_EDGES;        // self loop
  }
  float nrm = dinv[s] * dinv[d];
  v4f h = *(const v4f*)(H + (long long)s * D + c);
  float* o = acc + (long long)d * D + c;
  unsafeAtomicAdd(o + 0, h.x * nrm);
  unsafeAtomicAdd(o + 1, h.y * nrm);
  unsafeAtomicAdd(o + 2, h.z * nrm);
  unsafeAtomicAdd(o + 3, h.w * nrm);
}

template <int D>
__global__ void bias_relu_kernel(float* __restrict__ A,
                                 const float* __restrict__ b) {
  long long i = (long long)blockIdx.x * blockDim.x + threadIdx.x;
  if (i >= (long long)N_NODES * D) return;
  float v = A[i] + b[(int)(i & (D - 1))];  // D is a power of two
  A[i] = (v > 0.0f) ? v : 0.0f;
}

// -------------------------------------------------------------- classifier
__global__ void classifier_kernel(const float* __restrict__ X,
                                  const float* __restrict__ CW,
                                  const float* __restrict__ CB,
                                  float* __restrict__ out) {
  int i = blockIdx.x * blockDim.x + threadIdx.x;
  if (i >= N_NODES * NCLS) return;
  int n = i / NCLS, c = i % NCLS;
  const float* x = X + (long long)n * 16;
  float s = CB[c];
#pragma unroll
  for (int k = 0; k < 16; ++k) s += x[k] * CW[k * NCLS + c];
  out[i] = s;
}

// ------------------------------------------------------------------- driver
extern "C" void kernel_launch(void* const* d_in, const int* in_sizes, int n_in,
                              void* d_out, int out_size, void* d_ws,
                              size_t ws_size, hipStream_t stream) {
  (void)in_sizes; (void)n_in; (void)out_size; (void)ws_size;

  const float*     x0 = (const float*)d_in[0];
  const long long* ei = (const long long*)d_in[1];  // int64 edge_index [2,E]
  const float* W[6];
  const float* B[6];
  for (int i = 0; i < 6; ++i) {
    W[i] = (const float*)d_in[2 + i];
    B[i] = (const float*)d_in[8 + i];
  }
  const float* cw = (const float*)d_in[14];
  const float* cb = (const float*)d_in[15];

  // workspace layout (256B aligned): dinv | H | pingA | pingB
  char*  ws  = (char*)d_ws;
  size_t off = 0;
  auto alloc = [&](size_t bytes) -> void* {
    void* p = ws + off;
    off += (bytes + 255) & ~(size_t)255;
    return p;
  };
  float* dinv = (float*)alloc((size_t)N_NODES * 4);
  float* bufH = (float*)alloc((size_t)N_NODES * 256 * 4);
  float* bufA = (float*)alloc((size_t)N_NODES * 256 * 4);
  float* bufB = (float*)alloc((size_t)N_NODES * 256 * 4);

  const int TB = 256;
  auto blocks = [&](long long n) { return (unsigned)((n + TB - 1) / TB); };

  // symmetric normalization coefficients (recomputed every call; stateless)
  zero_f32<<<blocks(N_NODES), TB, 0, stream>>>(dinv, N_NODES);
  degree_kernel<<<blocks(N_EDGES + N_NODES), TB, 0, stream>>>(ei, dinv);
  rsqrt_kernel<<<blocks(N_NODES), TB, 0, stream>>>(dinv);

  const float* xin = x0;
  float* cur = bufA;
  float* nxt = bufB;

#define LAYER(DIN, DOUT, Wp, Bp)                                               \
  do {                                                                         \
    constexpr int waves = (N_NODES / 16) * ((DOUT) / 16);                      \
    gemm_wmma_f32<DIN, DOUT>                                                   \
        <<<(waves + 7) / 8, 256, 0, stream>>>(xin, (Wp), bufH);                \
    zero_f32<<<blocks((long long)N_NODES * (DOUT)), TB, 0, stream>>>(          \
        cur, (long long)N_NODES * (DOUT));                                     \
    aggregate_kernel<DOUT>                                                     \
        <<<blocks((long long)(N_EDGES + N_NODES) * ((DOUT) / 4)), TB, 0,       \
           stream>>>(bufH, ei, dinv, cur);                                     \
    bias_relu_kernel<DOUT>                                                     \
        <<<blocks((long long)N_NODES * (DOUT)), TB, 0, stream>>>(cur, (Bp));   \
    xin = cur;                                                                 \
    { float* t = cur; cur = nxt; nxt = t; }                                    \
  } while (0)

  LAYER(128, 16,  W[0], B[0]);
  LAYER(16,  64,  W[1], B[1]);
  LAYER(64,  256, W[2], B[2]);
  LAYER(256, 256, W[3], B[3]);
  LAYER(256, 64,  W[4], B[4]);
  LAYER(64,  16,  W[5], B[5]);
#undef LAYER

  classifier_kernel<<<blocks((long long)N_NODES * NCLS), TB, 0, stream>>>(
      xin, cw, cb, (float*)d_out);
}